// FCLSTMCell_24034636988552
// MI455X (gfx1250) — compile-verified
//
#include <hip/hip_runtime.h>
#include <hip/hip_bf16.h>

// ---------------------------------------------------------------------------
// FCLSTMCell for MI455X (gfx1250): fp32 WMMA GEMMs + fused LN/gate epilogue.
// B=256, IN=1024, H=1024, XD=16384, 4H=4096.
// GEMM: 64x64 block tile, 4 waves (128 thr), 32x32 wave tile (4 accumulators),
// K-pair-interleaved B layout in LDS so every WMMA fragment is one ds_load_b64.
// A tile staged with GLOBAL_LOAD_ASYNC_TO_LDS_B128 when the builtin exists.
// ---------------------------------------------------------------------------

typedef __attribute__((ext_vector_type(2))) float v2f;
typedef __attribute__((ext_vector_type(8))) float v8f;
typedef int v4i __attribute__((vector_size(16)));

#define BATCH   256
#define HID     1024
#define NGATE   4096   // 4*H
#define XDIM    16384
#define INDIM   1024

#define BM 64
#define BN 64
#define BK 64
#define ALD 68          // A LDS row stride (dwords): mult-of-4 -> 16B-aligned rows
#define BLD 160         // B LDS kpair stride (dwords): 128 data + 32 pad (bank skew)

#if __has_builtin(__builtin_amdgcn_global_load_async_to_lds_b128)
#define HAVE_ASYNC_LDS 1
#else
#define HAVE_ASYNC_LDS 0
#endif

__device__ __forceinline__ void async_copy16(const float* gsrc, float* ldst)
{
#if HAVE_ASYNC_LDS
    __builtin_amdgcn_global_load_async_to_lds_b128(
        (__attribute__((address_space(1))) v4i*)gsrc,
        (__attribute__((address_space(3))) v4i*)ldst,
        0, 0);
#else
    *(float4*)ldst = *(const float4*)gsrc;
#endif
}

__device__ __forceinline__ void wait_async_lds()
{
#if HAVE_ASYNC_LDS
#if __has_builtin(__builtin_amdgcn_s_wait_asynccnt)
    __builtin_amdgcn_s_wait_asynccnt(0);
#else
    asm volatile("s_wait_asynccnt 0" ::: "memory");
#endif
#endif
}

// C[M,N] = A[M,K] @ W[K,N]   (M = 256, N = 4096, K passed in; all mult of 64)
__launch_bounds__(128)
__global__ void gemm_wmma_f32(const float* __restrict__ A,
                              const float* __restrict__ W,
                              float* __restrict__ C,
                              int K, int N)
{
    __shared__ float As[BM * ALD];        // As[m][k], k contiguous
    __shared__ float Bi[(BK / 2) * BLD];  // Bi[kpair][n*2 + (k&1)], interleaved

    const int t    = threadIdx.x;
    const int lane = t & 31;
    const int wave = t >> 5;              // 4 waves
    const int ln   = lane & 15;           // 0..15 within half-wave
    const int koff = (lane >> 4) << 1;    // 0 (lanes 0-15) or 2 (lanes 16-31)

    const int waveM = (wave & 1) * 32;    // 2 waves along M, 32 rows each
    const int waveN = (wave >> 1) * 32;   // 2 waves along N, 32 cols each
    const int mBase = blockIdx.y * BM;
    const int nBase = blockIdx.x * BN;

    v8f acc00 = {}, acc01 = {}, acc10 = {}, acc11 = {};

    for (int k0 = 0; k0 < K; k0 += BK) {
        // --- stage A tile (BM x BK) into LDS, async DMA when available -----
#pragma unroll
        for (int it = 0; it < 8; ++it) {
            int c   = t + it * 128;          // 0..1023 float4 units
            int row = c >> 4;                // 64 rows
            int kq  = (c & 15) << 2;         // 16 float4 per row
            async_copy16(A + (size_t)(mBase + row) * K + k0 + kq,
                         &As[row * ALD + kq]);
        }
        // --- stage B tile (BK x BN) into LDS, K-pair interleaved -----------
#pragma unroll
        for (int it = 0; it < 4; ++it) {
            int c     = t + it * 128;        // 0..511 units (8 floats each)
            int kpair = c >> 4;              // 32 kpairs
            int n     = (c & 15) << 2;       // 16 4-col groups
            const float4 r0 = *(const float4*)(W + (size_t)(k0 + 2 * kpair)     * N + nBase + n);
            const float4 r1 = *(const float4*)(W + (size_t)(k0 + 2 * kpair + 1) * N + nBase + n);
            float* d = &Bi[kpair * BLD + n * 2];
            *(float4*)(d)     = make_float4(r0.x, r1.x, r0.y, r1.y);
            *(float4*)(d + 4) = make_float4(r0.z, r1.z, r0.w, r1.w);
        }
        wait_async_lds();
        __syncthreads();

        // --- prefetch next K-chunk (global_prefetch_b8) --------------------
        if (k0 + BK < K) {
            int row = t >> 3;                // 0..15
            int q   = (t & 7) << 3;
            __builtin_prefetch(A + (size_t)(mBase + row) * K + (k0 + BK) + q, 0, 0);
            __builtin_prefetch(W + (size_t)(k0 + BK + row) * N + nBase + q, 0, 0);
        }

        // --- compute: 16 k-steps x 4 WMMAs per wave ------------------------
#pragma unroll
        for (int kk = 0; kk < BK; kk += 4) {
            const int krow = kk + koff;      // even
            const int kp   = krow >> 1;
            // A fragments: rows [waveM..+15] and [waveM+16..+31]
            const v2f a0 = *(const v2f*)(&As[(waveM + ln)      * ALD + krow]);
            const v2f a1 = *(const v2f*)(&As[(waveM + 16 + ln) * ALD + krow]);
            // B fragments: single b64 each thanks to interleaved layout
            const v2f b0 = *(const v2f*)(&Bi[kp * BLD + (waveN + ln)      * 2]);
            const v2f b1 = *(const v2f*)(&Bi[kp * BLD + (waveN + 16 + ln) * 2]);

            acc00 = __builtin_amdgcn_wmma_f32_16x16x4_f32(
                false, a0, false, b0, (short)0, acc00, false, false);
            acc01 = __builtin_amdgcn_wmma_f32_16x16x4_f32(
                false, a0, false, b1, (short)0, acc01, false, false);
            acc10 = __builtin_amdgcn_wmma_f32_16x16x4_f32(
                false, a1, false, b0, (short)0, acc10, false, false);
            acc11 = __builtin_amdgcn_wmma_f32_16x16x4_f32(
                false, a1, false, b1, (short)0, acc11, false, false);
        }
        __syncthreads();
    }

    // --- epilogue: C/D layout -> VGPR r holds (M=r | M=r+8), N=lane%16 -----
    const int rowHalf = (lane >> 4) << 3;   // 0 or 8
#pragma unroll
    for (int r = 0; r < 8; ++r) {
        int row0 = mBase + waveM + rowHalf + r;
        int row1 = row0 + 16;
        C[(size_t)row0 * N + nBase + waveN + ln]      = acc00[r];
        C[(size_t)row0 * N + nBase + waveN + 16 + ln] = acc01[r];
        C[(size_t)row1 * N + nBase + waveN + ln]      = acc10[r];
        C[(size_t)row1 * N + nBase + waveN + 16 + ln] = acc11[r];
    }
}

// ---------------------------------------------------------------------------
// Fused bias + LayerNorm (both branches) + LSTM gating.
// One block per batch row; 256 threads; each thread owns j = t + 256*i,
// i = 0..15, of both 4096-wide pre-activation rows. Gate slices for hidden
// unit k = t + 256*i0 (i0<4) live on the same thread at reg index i0 + 4*g.
// ---------------------------------------------------------------------------
__device__ __forceinline__ float sigmoidf_(float x) {
    return 1.0f / (1.0f + __expf(-x));
}

__launch_bounds__(256)
__global__ void ln_gate_kernel(const float* __restrict__ zx,
                               const float* __restrict__ zh,
                               const float* __restrict__ c_t,
                               const float* __restrict__ b_h,
                               const float* __restrict__ gamma_h,
                               const float* __restrict__ beta_h,
                               const float* __restrict__ b_x,
                               const float* __restrict__ gamma_x,
                               const float* __restrict__ beta_x,
                               const float* __restrict__ ct_w,
                               const float* __restrict__ oc_w,
                               float* __restrict__ h_out,
                               float* __restrict__ c_out)
{
    const int b = blockIdx.x;
    const int t = threadIdx.x;

    float rh[16], rx[16];
    float sh = 0.f, sqh = 0.f, sx = 0.f, sqx = 0.f;

#pragma unroll
    for (int i = 0; i < 16; ++i) {
        int j = t + (i << 8);
        float vh = zh[(size_t)b * NGATE + j] + b_h[j];
        float vx = zx[(size_t)b * NGATE + j] + b_x[j];
        rh[i] = vh; rx[i] = vx;
        sh += vh; sqh += vh * vh;
        sx += vx; sqx += vx * vx;
    }

    __shared__ float4 red[256];
    red[t].x = sh; red[t].y = sqh; red[t].z = sx; red[t].w = sqx;
    __syncthreads();
    for (int off = 128; off > 0; off >>= 1) {
        if (t < off) {
            red[t].x += red[t + off].x;
            red[t].y += red[t + off].y;
            red[t].z += red[t + off].z;
            red[t].w += red[t + off].w;
        }
        __syncthreads();
    }
    const float4 tot = red[0];
    const float inv_n  = 1.0f / (float)NGATE;
    const float mu_h   = tot.x * inv_n;
    const float var_h  = tot.y * inv_n - mu_h * mu_h;
    const float rstd_h = __frsqrt_rn(var_h + 1e-5f);
    const float mu_x   = tot.z * inv_n;
    const float var_x  = tot.w * inv_n - mu_x * mu_x;
    const float rstd_x = __frsqrt_rn(var_x + 1e-5f);

#pragma unroll
    for (int i0 = 0; i0 < 4; ++i0) {
        const int k = t + (i0 << 8);
        // h branch gate order: i, f, g, o
        float ih = (rh[i0 + 0]  - mu_h) * rstd_h * gamma_h[k]        + beta_h[k];
        float fh = (rh[i0 + 4]  - mu_h) * rstd_h * gamma_h[k + 1024] + beta_h[k + 1024];
        float gh = (rh[i0 + 8]  - mu_h) * rstd_h * gamma_h[k + 2048] + beta_h[k + 2048];
        float oh = (rh[i0 + 12] - mu_h) * rstd_h * gamma_h[k + 3072] + beta_h[k + 3072];
        // x branch gate order: i, g, f, o
        float ix = (rx[i0 + 0]  - mu_x) * rstd_x * gamma_x[k]        + beta_x[k];
        float gx = (rx[i0 + 4]  - mu_x) * rstd_x * gamma_x[k + 1024] + beta_x[k + 1024];
        float fx = (rx[i0 + 8]  - mu_x) * rstd_x * gamma_x[k + 2048] + beta_x[k + 2048];
        float ox = (rx[i0 + 12] - mu_x) * rstd_x * gamma_x[k + 3072] + beta_x[k + 3072];

        const float c  = c_t[(size_t)b * HID + k];
        const float ic = c * ct_w[k];
        const float fc = c * ct_w[HID + k];

        const float ii = sigmoidf_(ih + ic + ix);
        const float ff = sigmoidf_(fh + fc + fx + 1.0f);   // FORGET_BIAS
        const float gg = tanhf(gh + gx);
        const float cn = ff * c + ii * gg;
        const float oo = oh + ox + cn * oc_w[k];
        const float hn = sigmoidf_(oo) * tanhf(cn);

        h_out[(size_t)b * HID + k] = hn;
        c_out[(size_t)b * HID + k] = cn;
    }
}

// ---------------------------------------------------------------------------
extern "C" void kernel_launch(void* const* d_in, const int* in_sizes, int n_in,
                              void* d_out, int out_size, void* d_ws, size_t ws_size,
                              hipStream_t stream)
{
    const float* x_t     = (const float*)d_in[0];
    const float* h_t     = (const float*)d_in[1];
    const float* c_t     = (const float*)d_in[2];
    const float* W_h     = (const float*)d_in[3];
    const float* b_h     = (const float*)d_in[4];
    const float* gamma_h = (const float*)d_in[5];
    const float* beta_h  = (const float*)d_in[6];
    const float* W_x     = (const float*)d_in[7];
    const float* b_x     = (const float*)d_in[8];
    const float* gamma_x = (const float*)d_in[9];
    const float* beta_x  = (const float*)d_in[10];
    const float* ct_w    = (const float*)d_in[11];
    const float* oc_w    = (const float*)d_in[12];

    float* out   = (float*)d_out;                 // h_new (256*1024) || c_new (256*1024)
    float* zx    = (float*)d_ws;                  // 256*4096 floats
    float* zh    = zx + (size_t)BATCH * NGATE;    // 256*4096 floats

    dim3 grid(NGATE / BN, BATCH / BM);            // (64, 4)
    gemm_wmma_f32<<<grid, 128, 0, stream>>>(x_t, W_x, zx, XDIM, NGATE);
    gemm_wmma_f32<<<grid, 128, 0, stream>>>(h_t, W_h, zh, INDIM, NGATE);

    ln_gate_kernel<<<BATCH, 256, 0, stream>>>(zx, zh, c_t,
                                              b_h, gamma_h, beta_h,
                                              b_x, gamma_x, beta_x,
                                              ct_w, oc_w,
                                              out, out + (size_t)BATCH * HID);
}